// GATEncoder_8735963480509
// MI455X (gfx1250) — compile-verified
//
#include <hip/hip_runtime.h>
#include <math.h>

typedef __attribute__((ext_vector_type(2))) float v2f;
typedef __attribute__((ext_vector_type(8))) float v8f;

#define HID 128
#define HEADS 4
#define CH 32
#define NEG_SLOPE 0.2f

// ---------------------------------------------------------------------------
// float atomic max via integer ordering trick (buffer initialized to -inf)
// ---------------------------------------------------------------------------
__device__ __forceinline__ void atomicMaxF(float* addr, float val) {
  int iv = __float_as_int(val);
  if (iv >= 0) {
    atomicMax((int*)addr, iv);
  } else {
    atomicMin((unsigned int*)addr, (unsigned int)iv);
  }
}

__device__ __forceinline__ void atomAddF(float* addr, float val) {
  // hardware global_atomic_add_f32 (no CAS loop)
  unsafeAtomicAdd(addr, val);
}

// ---------------------------------------------------------------------------
// Kernel 1: H[N,128] = X[N,128] @ W[128,128] via V_WMMA_F32_16X16X4_F32.
// One wave per 16-row strip; 8 accumulator tiles cover all 128 output cols.
// A 16x4 layout: lane<16 -> {K=k0,k0+1}, lane>=16 -> {K=k0+2,k0+3}, M=lane&15
// B 4x16 layout: lane<16 -> rows K=k0,k0+1; lane>=16 -> rows K=k0+2,k0+3; N=lane&15
// C/D: VGPR v -> row m0 + v + 8*(lane>=16), col = n0 + (lane&15)
// ---------------------------------------------------------------------------
__global__ __launch_bounds__(128) void k_gemm(const float* __restrict__ X,
                                              const float* __restrict__ W,
                                              float* __restrict__ H, int nrows) {
  const int wave = blockIdx.x * 4 + (threadIdx.x >> 5);
  const int lane = threadIdx.x & 31;
  const int m0 = wave << 4;
  if (m0 >= nrows) return;           // wave-uniform: EXEC stays all-ones for WMMA
  const int half = lane >> 4;        // 0 or 1
  const int l16 = lane & 15;

  const float* xrow = X + (size_t)(m0 + l16) * HID;
  v8f acc[8] = {};

  for (int k0 = 0; k0 < HID; k0 += 4) {
    v2f a;
    a.x = xrow[k0 + 2 * half];
    a.y = xrow[k0 + 2 * half + 1];
    const float* wr0 = W + (size_t)(k0 + 2 * half) * HID + l16;
    const float* wr1 = wr0 + HID;
#pragma unroll
    for (int t = 0; t < 8; ++t) {
      v2f b;
      b.x = wr0[t * 16];
      b.y = wr1[t * 16];
      acc[t] = __builtin_amdgcn_wmma_f32_16x16x4_f32(
          false, a, false, b, (short)0, acc[t], false, false);
    }
  }

#pragma unroll
  for (int v = 0; v < 8; ++v) {
    float* orow = H + (size_t)(m0 + v + half * 8) * HID + l16;
#pragma unroll
    for (int t = 0; t < 8; ++t) orow[t * 16] = acc[t][v];
  }
}

// ---------------------------------------------------------------------------
// Kernel 2: per-(node,head) attention dots; also init m=-inf, denom=0, agg=0
// ---------------------------------------------------------------------------
__global__ __launch_bounds__(256) void k_att_prep(
    const float* __restrict__ H, const float* __restrict__ att_src,
    const float* __restrict__ att_dst, float* __restrict__ a_s,
    float* __restrict__ a_d, float* __restrict__ m, float* __restrict__ den,
    float* __restrict__ agg, int n_nodes) {
  int idx = blockIdx.x * blockDim.x + threadIdx.x;  // n*HEADS + head
  if (idx >= n_nodes * HEADS) return;
  int head = idx & (HEADS - 1);
  int n = idx >> 2;
  const float* hrow = H + (size_t)n * HID + head * CH;
  const float* as = att_src + head * CH;
  const float* ad = att_dst + head * CH;
  float* aggp = agg + (size_t)n * HID + head * CH;
  float ss = 0.f, sd = 0.f;
#pragma unroll 8
  for (int c = 0; c < CH; ++c) {
    float hv = hrow[c];
    ss += hv * as[c];
    sd += hv * ad[c];
    aggp[c] = 0.f;
  }
  a_s[idx] = ss;
  a_d[idx] = sd;
  m[idx] = -__builtin_inff();
  den[idx] = 0.f;
}

__device__ __forceinline__ void edge_ids(const int* __restrict__ ei, int i,
                                         int E_, int* s, int* d) {
  if (i < E_) { *s = ei[i]; *d = ei[E_ + i]; }
  else        { *s = *d = i - E_; }      // self-loops appended
}

// ---------------------------------------------------------------------------
// Kernel 3: segment max of leaky_relu(a_s[src]+a_d[dst]) into m[dst]
// ---------------------------------------------------------------------------
__global__ __launch_bounds__(256) void k_edge_max(
    const int* __restrict__ ei, const float* __restrict__ a_s,
    const float* __restrict__ a_d, float* __restrict__ m, int E_, int N_) {
  int i = blockIdx.x * blockDim.x + threadIdx.x;
  if (i >= E_ + N_) return;
  int s, d;
  edge_ids(ei, i, E_, &s, &d);
  float4 es = *(const float4*)(a_s + (size_t)s * 4);
  float4 ed = *(const float4*)(a_d + (size_t)d * 4);
  float e[4] = {es.x + ed.x, es.y + ed.y, es.z + ed.z, es.w + ed.w};
  float* mp = m + (size_t)d * 4;
#pragma unroll
  for (int h = 0; h < 4; ++h) {
    float v = e[h] >= 0.f ? e[h] : NEG_SLOPE * e[h];
    atomicMaxF(mp + h, v);
  }
}

// ---------------------------------------------------------------------------
// Kernel 4: denom[dst] += exp(e - m[dst])
// ---------------------------------------------------------------------------
__global__ __launch_bounds__(256) void k_edge_denom(
    const int* __restrict__ ei, const float* __restrict__ a_s,
    const float* __restrict__ a_d, const float* __restrict__ m,
    float* __restrict__ den, int E_, int N_) {
  int i = blockIdx.x * blockDim.x + threadIdx.x;
  if (i >= E_ + N_) return;
  int s, d;
  edge_ids(ei, i, E_, &s, &d);
  float4 es = *(const float4*)(a_s + (size_t)s * 4);
  float4 ed = *(const float4*)(a_d + (size_t)d * 4);
  float4 mm = *(const float4*)(m + (size_t)d * 4);
  float e[4] = {es.x + ed.x, es.y + ed.y, es.z + ed.z, es.w + ed.w};
  float mv[4] = {mm.x, mm.y, mm.z, mm.w};
  float* dp = den + (size_t)d * 4;
#pragma unroll
  for (int h = 0; h < 4; ++h) {
    float v = e[h] >= 0.f ? e[h] : NEG_SLOPE * e[h];
    atomAddF(dp + h, __expf(v - mv[h]));
  }
}

// ---------------------------------------------------------------------------
// Kernel 5: agg[dst] += alpha * h[src]; one wave32 per edge, 4 channels/lane
// ---------------------------------------------------------------------------
__global__ __launch_bounds__(256) void k_edge_aggr(
    const int* __restrict__ ei, const float* __restrict__ H,
    const float* __restrict__ a_s, const float* __restrict__ a_d,
    const float* __restrict__ m, const float* __restrict__ den,
    float* __restrict__ agg, int E_, int N_) {
  int wave = blockIdx.x * 8 + (threadIdx.x >> 5);
  int lane = threadIdx.x & 31;
  if (wave >= E_ + N_) return;
  int s, d;
  edge_ids(ei, wave, E_, &s, &d);
  int head = lane >> 3;  // 8 lanes (32 ch) per head
  float e = a_s[(size_t)s * 4 + head] + a_d[(size_t)d * 4 + head];
  e = e >= 0.f ? e : NEG_SLOPE * e;
  float alpha = __expf(e - m[(size_t)d * 4 + head]) /
                (den[(size_t)d * 4 + head] + 1e-16f);
  float4 hv = *(const float4*)(H + (size_t)s * HID + lane * 4);
  float* op = agg + (size_t)d * HID + lane * 4;
  atomAddF(op + 0, alpha * hv.x);
  atomAddF(op + 1, alpha * hv.y);
  atomAddF(op + 2, alpha * hv.z);
  atomAddF(op + 3, alpha * hv.w);
}

// ---------------------------------------------------------------------------
// Kernel 6: out = LayerNorm(ELU(agg + bias)) ; one wave32 per node
// ---------------------------------------------------------------------------
__global__ __launch_bounds__(256) void k_elu_ln(
    const float* __restrict__ agg, const float* __restrict__ bias,
    const float* __restrict__ gamma, const float* __restrict__ beta,
    float* __restrict__ out, int n_nodes) {
  int wave = blockIdx.x * 8 + (threadIdx.x >> 5);
  int lane = threadIdx.x & 31;
  if (wave >= n_nodes) return;
  float4 v = *(const float4*)(agg + (size_t)wave * HID + lane * 4);
  float4 b = *(const float4*)(bias + lane * 4);
  float ev[4] = {v.x + b.x, v.y + b.y, v.z + b.z, v.w + b.w};
  float s = 0.f, s2 = 0.f;
#pragma unroll
  for (int j = 0; j < 4; ++j) {
    float t = ev[j];
    t = t > 0.f ? t : expm1f(t);  // ELU
    ev[j] = t;
    s += t;
    s2 += t * t;
  }
#pragma unroll
  for (int off = 16; off >= 1; off >>= 1) {  // wave32 tree reduction
    s += __shfl_xor(s, off, 32);
    s2 += __shfl_xor(s2, off, 32);
  }
  float mu = s * (1.f / HID);
  float var = s2 * (1.f / HID) - mu * mu;
  float rstd = rsqrtf(var + 1e-5f);
  float4 g = *(const float4*)(gamma + lane * 4);
  float4 be = *(const float4*)(beta + lane * 4);
  float4 o;
  o.x = (ev[0] - mu) * rstd * g.x + be.x;
  o.y = (ev[1] - mu) * rstd * g.y + be.y;
  o.z = (ev[2] - mu) * rstd * g.z + be.z;
  o.w = (ev[3] - mu) * rstd * g.w + be.w;
  *(float4*)(out + (size_t)wave * HID + lane * 4) = o;
}

// ---------------------------------------------------------------------------
extern "C" void kernel_launch(void* const* d_in, const int* in_sizes, int n_in,
                              void* d_out, int out_size, void* d_ws,
                              size_t ws_size, hipStream_t stream) {
  const float* x   = (const float*)d_in[0];
  const int*   ei  = (const int*)d_in[1];
  const float* W1  = (const float*)d_in[2];
  const float* as1 = (const float*)d_in[3];
  const float* ad1 = (const float*)d_in[4];
  const float* b1  = (const float*)d_in[5];
  const float* W2  = (const float*)d_in[6];
  const float* as2 = (const float*)d_in[7];
  const float* ad2 = (const float*)d_in[8];
  const float* b2  = (const float*)d_in[9];
  const float* g1  = (const float*)d_in[10];
  const float* be1 = (const float*)d_in[11];
  const float* g2  = (const float*)d_in[12];
  const float* be2 = (const float*)d_in[13];
  float* out = (float*)d_out;

  const int N_ = in_sizes[0] / HID;
  const int E_ = in_sizes[1] / 2;
  const int TOT = E_ + N_;

  char* ws = (char*)d_ws;
  size_t off = 0;
  auto wsalloc = [&](size_t bytes) -> float* {
    float* p = (float*)(ws + off);
    off = (off + bytes + 255) & ~(size_t)255;
    return p;
  };
  float* h   = wsalloc((size_t)N_ * HID * sizeof(float));
  float* agg = wsalloc((size_t)N_ * HID * sizeof(float));
  float* a_s = wsalloc((size_t)N_ * HEADS * sizeof(float));
  float* a_d = wsalloc((size_t)N_ * HEADS * sizeof(float));
  float* mb  = wsalloc((size_t)N_ * HEADS * sizeof(float));
  float* den = wsalloc((size_t)N_ * HEADS * sizeof(float));

  auto run_layer = [&](const float* xin, const float* W, const float* as,
                       const float* ad, const float* bias, const float* gamma,
                       const float* beta, float* dst) {
    int gwaves = (N_ + 15) / 16;
    k_gemm<<<(gwaves + 3) / 4, 128, 0, stream>>>(xin, W, h, N_);
    k_att_prep<<<(N_ * HEADS + 255) / 256, 256, 0, stream>>>(
        h, as, ad, a_s, a_d, mb, den, agg, N_);
    k_edge_max<<<(TOT + 255) / 256, 256, 0, stream>>>(ei, a_s, a_d, mb, E_, N_);
    k_edge_denom<<<(TOT + 255) / 256, 256, 0, stream>>>(ei, a_s, a_d, mb, den,
                                                        E_, N_);
    k_edge_aggr<<<(TOT + 7) / 8, 256, 0, stream>>>(ei, h, a_s, a_d, mb, den,
                                                   agg, E_, N_);
    k_elu_ln<<<(N_ + 7) / 8, 256, 0, stream>>>(agg, bias, gamma, beta, dst, N_);
  };

  run_layer(x,   W1, as1, ad1, b1, g1, be1, out);  // layer-1 -> d_out (temp)
  run_layer(out, W2, as2, ad2, b2, g2, be2, out);  // layer-2 -> d_out (final)
}